// ComplementConstraint_14448269984499
// MI455X (gfx1250) — compile-verified
//
#include <hip/hip_runtime.h>
#include <hip/hip_bf16.h>

// ComplementConstraint: per-row leave-one-out logsumexp.
//   lse_b   = logsumexp(x[b,:])
//   out[b,k]= -(lse_b + log(1 - exp(x[b,k] - lse_b)))
//
// B=16384 rows, C=1000 cols, fp32 in/out. Memory-bound: ~131 MB total
// traffic -> ~5.6 us floor at 23.3 TB/s. One wave32 per row; rows streamed
// to LDS with CDNA5 async global->LDS loads, double-buffered per wave.

typedef float v4f __attribute__((ext_vector_type(4)));

#define NBLOCKS        1024
#define WAVES_PER_BLK  8
#define ROW_FLOATS     1000
#define ROW_BYTES      4000u
#define CHUNKS         250      // float4 chunks per row
#define LDS_ROW_F      1024     // padded row (floats) in LDS
#define NROWS          16384

// Issue one row (8 x b128 async loads) from global into LDS.
// Tail chunks (c >= 250) are clamped to the row base so all 8 issues run
// with full EXEC (deterministic ASYNCcnt) and never read out of bounds.
__device__ __forceinline__ void issue_row_async(const float* __restrict__ gbase,
                                                unsigned row_byte_off,
                                                unsigned lds_byte_base,
                                                int lane) {
#pragma unroll
    for (int k = 0; k < 8; ++k) {
        unsigned c     = (unsigned)(lane + 32 * k);
        unsigned gc    = (c < CHUNKS) ? c : 0u;          // clamp OOB tail
        unsigned goff  = row_byte_off + gc * 16u;        // 32-bit GVS offset
        unsigned laddr = lds_byte_base + c * 16u;        // LDS byte address
        asm volatile("global_load_async_to_lds_b128 %0, %1, %2"
                     :
                     : "v"(laddr), "v"(goff), "s"(gbase)
                     : "memory");
    }
}

__device__ __forceinline__ float wave_reduce_max(float v) {
#pragma unroll
    for (int off = 16; off > 0; off >>= 1)
        v = fmaxf(v, __shfl_xor(v, off, 32));
    return v;
}

__device__ __forceinline__ float wave_reduce_sum(float v) {
#pragma unroll
    for (int off = 16; off > 0; off >>= 1)
        v += __shfl_xor(v, off, 32);
    return v;
}

__device__ __forceinline__ void process_row(const float* __restrict__ lrow,
                                            float* __restrict__ orow,
                                            int lane) {
    const v4f* l4 = reinterpret_cast<const v4f*>(lrow);
    v4f v[8];
#pragma unroll
    for (int k = 0; k < 8; ++k)
        v[k] = l4[lane + 32 * k];               // ds_load_b128 x8

    // Mask the invalid tail of chunk 7 (c = 224+lane >= 250 for lane >= 26):
    // -inf is neutral for max and exp(-inf - m) == 0 for the sum.
    const float NEG_INF = -__builtin_inff();
    if (lane >= 26) { v[7].x = NEG_INF; v[7].y = NEG_INF; v[7].z = NEG_INF; v[7].w = NEG_INF; }

    // 1) row max
    float m = NEG_INF;
#pragma unroll
    for (int k = 0; k < 8; ++k)
        m = fmaxf(m, fmaxf(fmaxf(v[k].x, v[k].y), fmaxf(v[k].z, v[k].w)));
    m = wave_reduce_max(m);

    // 2) e_k = exp(x_k - m) computed ONCE, kept in regs; accumulate S
    float s = 0.0f;
#pragma unroll
    for (int k = 0; k < 8; ++k) {
        v4f e;
        e.x = __expf(v[k].x - m);
        e.y = __expf(v[k].y - m);
        e.z = __expf(v[k].z - m);
        e.w = __expf(v[k].w - m);
        v[k] = e;
        s += (e.x + e.y) + (e.z + e.w);
    }
    s = wave_reduce_sum(s);

    const float lse = m + __logf(s);
    const float rS  = __builtin_amdgcn_rcpf(s);

    // 3) out_k = -(lse + log(1 - e_k/S))
    v4f* o4 = reinterpret_cast<v4f*>(orow);
#pragma unroll
    for (int k = 0; k < 8; ++k) {
        int c = lane + 32 * k;
        if (c < CHUNKS) {
            v4f o;
            o.x = -(lse + __logf(1.0f - v[k].x * rS));
            o.y = -(lse + __logf(1.0f - v[k].y * rS));
            o.z = -(lse + __logf(1.0f - v[k].z * rS));
            o.w = -(lse + __logf(1.0f - v[k].w * rS));
            o4[c] = o;                           // global_store_b128
        }
    }
}

__global__ __launch_bounds__(256)
void ComplementConstraint_kernel(const float* __restrict__ in,
                                 float* __restrict__ out) {
    // 64 KB: 8 waves x 2 buffers x 4 KB
    __shared__ __align__(16) float smem[WAVES_PER_BLK * 2 * LDS_ROW_F];

    const int lane = threadIdx.x & 31;
    const int wave = threadIdx.x >> 5;

    // Low 32 bits of the generic shared pointer == LDS byte offset.
    const unsigned lds_base = (unsigned)(size_t)(&smem[0]);
    const unsigned lb0 = lds_base + (unsigned)((wave * 2 + 0) * LDS_ROW_F * 4);
    const unsigned lb1 = lds_base + (unsigned)((wave * 2 + 1) * LDS_ROW_F * 4);

    const int wslot = (int)blockIdx.x * WAVES_PER_BLK + wave;
    const int r0 = wslot;                                   // rows 0..8191
    const int r1 = wslot + NBLOCKS * WAVES_PER_BLK;         // rows 8192..16383

    // Prefetch both rows (double buffered): 16 async ops in flight max.
    issue_row_async(in, (unsigned)r0 * ROW_BYTES, lb0, lane);
    issue_row_async(in, (unsigned)r1 * ROW_BYTES, lb1, lane);

    // Row 0's 8 loads done when ASYNCcnt <= 8 (async loads retire in order).
    asm volatile("s_wait_asynccnt 8" ::: "memory");
    process_row(smem + (wave * 2 + 0) * LDS_ROW_F,
                out + (size_t)r0 * ROW_FLOATS, lane);

    asm volatile("s_wait_asynccnt 0" ::: "memory");
    process_row(smem + (wave * 2 + 1) * LDS_ROW_F,
                out + (size_t)r1 * ROW_FLOATS, lane);
}

extern "C" void kernel_launch(void* const* d_in, const int* in_sizes, int n_in,
                              void* d_out, int out_size, void* d_ws, size_t ws_size,
                              hipStream_t stream) {
    const float* in = (const float*)d_in[0];
    float* out      = (float*)d_out;
    (void)in_sizes; (void)n_in; (void)out_size; (void)d_ws; (void)ws_size;

    hipLaunchKernelGGL(ComplementConstraint_kernel,
                       dim3(NBLOCKS), dim3(WAVES_PER_BLK * 32), 0, stream,
                       in, out);
}